// kernel_generated_0_21887153341171
// MI455X (gfx1250) — compile-verified
//
#include <hip/hip_runtime.h>

// CDNA5 / gfx1250, wave32. Single fused kernel:
//   out(64x3136) = A'(64x384) @ B'(384x3136)
// A'[i, h*192 + tap*64 + j] = w1[i,h] * w2[i,j,tap]
// B'[h*192 + tap*64 + j, (n,l)] = x[j, (n-1-h)%56, l+tap-1]  (0 outside col pad)
// V_WMMA_F32_16X16X4_F32, 96 WMMAs per 16x16 output tile, full f32 precision.
//
// K is ordered tap-major so the inner loop has no div/mod; out-of-pad taps use
// clamped addresses + v_cndmask (no exec-masked branches, EXEC stays all-ones).

typedef __attribute__((ext_vector_type(2))) float v2f;
typedef __attribute__((ext_vector_type(8))) float v8f;

#define HW      56
#define SPATIAL (HW * HW)   // 3136
#define KDIM    192         // 64 channels * 3 taps (per shift term)

__global__ __launch_bounds__(256)
void conv_shift_fold_wmma(const float* __restrict__ x,    // (64,56,56)
                          const float* __restrict__ w1,   // (64,2)
                          const float* __restrict__ w2,   // (64,64,3)
                          float* __restrict__ out)        // (64,56,56)
{
    const int lane = threadIdx.x & 31;
    const int wave = threadIdx.x >> 5;
    const int half = lane >> 4;      // 0: lanes 0-15, 1: lanes 16-31
    const int ln   = lane & 15;

    const int tile  = blockIdx.x * 8 + wave;   // 0..783
    const int i0 = (tile & 3) * 16;            // channel tile base
    const int s0 = (tile >> 2) * 16;           // spatial tile base

    // ---- Per-lane B/D column: output spatial position
    const int sOut = s0 + ln;
    const int nOut = sOut / HW;      // output row (roll axis)
    const int lCol = sOut % HW;      // output column (conv axis)

    // ---- Per-lane A row: output channel (same for both lane halves)
    const int   iRow   = i0 + ln;
    const float w1s0   = w1[iRow * 2 + 0];   // weight of shift-by-1 term
    const float w1s1   = w1[iRow * 2 + 1];   // weight of shift-by-2 term
    const float* w2row = w2 + iRow * KDIM;   // w2[iRow, j, tap] = w2row[j*3+tap]

    // ---- Source rows for the two shift terms: (nOut-1-h) mod 56
    int nSrc0 = nOut - 1; if (nSrc0 < 0) nSrc0 += HW;
    int nSrc1 = nOut - 2; if (nSrc1 < 0) nSrc1 += HW;
    const float* xr0 = x + nSrc0 * HW;   // + j*SPATIAL + column
    const float* xr1 = x + nSrc1 * HW;

    // ---- Column-pad predicates (hoisted; tap reads column lCol + tap - 1)
    const bool vTap0 = (lCol >= 1);        // tap 0
    const bool vTap2 = (lCol <= HW - 2);   // tap 2; tap 1 always valid

    v8f acc = {0.f, 0.f, 0.f, 0.f, 0.f, 0.f, 0.f, 0.f};

    for (int tap = 0; tap < 3; ++tap) {
        const bool valid = (tap == 0) ? vTap0 : ((tap == 2) ? vTap2 : true);
        const int  mOff  = lCol + tap - 1;
        const int  mSafe = valid ? mOff : lCol;    // always in-bounds address

        // Lane covers channels j = jA, jA+1 per K-chunk (A K=2*half+{0,1})
        const float* p0 = xr0 + 2 * half * SPATIAL + mSafe;
        const float* p1 = xr1 + 2 * half * SPATIAL + mSafe;
        const float* pw = w2row + 2 * half * 3 + tap;

#pragma unroll 4
        for (int j0 = 0; j0 < 64; j0 += 4) {
            // raw w2 pair (shared by both shift terms)
            const float wx = pw[0];
            const float wy = pw[3];

            // B pairs for shift-1 and shift-2 rows (unconditional loads,
            // value masked to zero outside the column pad)
            const float b0x = p0[0],       b0y = p0[SPATIAL];
            const float b1x = p1[0],       b1y = p1[SPATIAL];

            v2f b0; b0.x = valid ? b0x : 0.f;  b0.y = valid ? b0y : 0.f;
            v2f b1; b1.x = valid ? b1x : 0.f;  b1.y = valid ? b1y : 0.f;
            v2f a0; a0.x = w1s0 * wx;          a0.y = w1s0 * wy;
            v2f a1; a1.x = w1s1 * wx;          a1.y = w1s1 * wy;

            acc = __builtin_amdgcn_wmma_f32_16x16x4_f32(
                    false, a0, false, b0, (short)0, acc, false, false);
            acc = __builtin_amdgcn_wmma_f32_16x16x4_f32(
                    false, a1, false, b1, (short)0, acc, false, false);

            p0 += 4 * SPATIAL;
            p1 += 4 * SPATIAL;
            pw += 12;              // 4 channels * 3 taps
        }
    }

    // ---- Store D: lane(half,ln) VGPR v -> out[i0 + v + 8*half, sOut]
#pragma unroll
    for (int v = 0; v < 8; ++v) {
        out[(i0 + v + 8 * half) * SPATIAL + sOut] = acc[v];
    }
}

extern "C" void kernel_launch(void* const* d_in, const int* in_sizes, int n_in,
                              void* d_out, int out_size, void* d_ws, size_t ws_size,
                              hipStream_t stream) {
    const float* x  = (const float*)d_in[0];   // (1,64,56,56)
    const float* w1 = (const float*)d_in[1];   // (64,2)
    const float* w2 = (const float*)d_in[2];   // (64,64,3)
    float* out = (float*)d_out;                // (1,64,56,56)

    // 784 16x16 output tiles, 1 wave32 per tile, 8 waves per 256-thread block
    dim3 grid(98), block(256);
    hipLaunchKernelGGL(conv_shift_fold_wmma, grid, block, 0, stream, x, w1, w2, out);
}